// Sampler_40836549050652
// MI455X (gfx1250) — compile-verified
//
#include <hip/hip_runtime.h>
#include <hip/hip_bf16.h>

#define BDIM    512          // 16 wave32 per row-block
#define CH      2048         // floats per async chunk (= BDIM * 4)
#define DEPTH   16           // async ring depth: 15 chunks (120KB) in flight
#define NBINS   4096         // 12-bit key histogram
#define NCOARSE 512          // 8 fine bins per coarse bin
#define CANDMAX 1024         // candidate buffer (power of 2 for bitonic)
#define KMAX    64

// Monotone float -> uint key: larger float => larger key.
__device__ __forceinline__ uint32_t mono_key(float f) {
  uint32_t u = __float_as_uint(f);
  return (u & 0x80000000u) ? ~u : (u | 0x80000000u);
}
__device__ __forceinline__ float mono_inv(uint32_t k) {
  uint32_t u = (k & 0x80000000u) ? (k & 0x7FFFFFFFu) : ~k;
  return __uint_as_float(u);
}
// Deterministic per-row uniform in [0,1) (splitmix-style, seed 42).
__device__ __forceinline__ float u01(uint32_t row) {
  unsigned long long z =
      ((unsigned long long)(row + 1u) * 0x9E3779B97F4A7C15ull) ^
      (42ull * 0xBF58476D1CE4E5B9ull);
  z ^= z >> 31; z *= 0x94D049BB133111EBull; z ^= z >> 29;
  return (float)(z >> 40) * (1.0f / 16777216.0f);
}

__global__ __launch_bounds__(BDIM)
void Sampler_40836549050652_kernel(const float* __restrict__ logits,
                                   const int* __restrict__ kptr,
                                   float* __restrict__ out,
                                   int B, int V) {
  __shared__ float              stage[DEPTH][CH];   // 128KB async staging ring
  __shared__ uint32_t           hist[NBINS];        // 16KB
  __shared__ uint32_t           coarse[NCOARSE];
  __shared__ uint32_t           scan[NCOARSE];
  __shared__ unsigned long long cand[CANDMAX];      // 8KB
  __shared__ uint32_t           ncand;
  __shared__ int                sC;
  __shared__ uint32_t           sAbove;
  __shared__ int                sTbin;

  const int row = blockIdx.x;
  const int tid = threadIdx.x;
  const float* __restrict__ rp = logits + (size_t)row * (size_t)V;

  int kk = *kptr;
  if (kk < 1) kk = 1;
  if (kk > KMAX) kk = KMAX;

  for (int i = tid; i < NBINS; i += BDIM) hist[i] = 0u;
  if (tid == 0) { ncand = 0u; sC = NCOARSE - 1; sAbove = 0u; sTbin = 0; }
  __syncthreads();

  // ---- Pass 1: histogram; deep async global->LDS pipeline (ASYNCcnt) -----
  // Each thread consumes exactly the 16B it issued itself, so the only
  // ordering needed is the per-wave s_wait_asynccnt (async loads complete
  // in order within a wave). No block barriers in the streaming loop.
  const int nch = (V + CH - 1) / CH;

  auto issue = [&](int c) {
    const int base = c * CH;
    int elems = V - base; if (elems > CH) elems = CH;
    int off = tid * 4;
    if (off > elems - 4) off = elems - 4;      // clamp: uniform issue per wave
    unsigned long long ga = (unsigned long long)(rp + base + off);
    uint32_t la = (uint32_t)(unsigned long long)&stage[c % DEPTH][off];
    asm volatile("global_load_async_to_lds_b128 %0, %1, off"
                 :: "v"(la), "v"(ga) : "memory");
  };

  for (int c = 0; c < DEPTH - 1 && c < nch; ++c) issue(c);  // prologue
  for (int c = 0; c < nch; ++c) {
    if (c + DEPTH - 1 < nch) {
      issue(c + DEPTH - 1);
      asm volatile("s_wait_asynccnt 0xf" ::: "memory");  // <=15: chunk c landed
    } else {
      asm volatile("s_wait_asynccnt 0x0" ::: "memory");  // tail drain
    }
    const int base = c * CH;
    int elems = V - base; if (elems > CH) elems = CH;
    const int off = tid * 4;
    if (off < elems) {
      float4 x = *(const float4*)&stage[c % DEPTH][off];  // one b128 LDS read
      float vv[4] = {x.x, x.y, x.z, x.w};
#pragma unroll
      for (int e = 0; e < 4; ++e)
        atomicAdd(&hist[mono_key(vv[e]) >> 20], 1u);
    }
  }
  __syncthreads();

  // ---- Threshold: parallel suffix-scan over coarse bins ------------------
  {
    uint32_t s = 0u;
#pragma unroll
    for (int i = 0; i < 8; ++i) s += hist[tid * 8 + i];
    coarse[tid] = s;
    scan[tid] = s;
  }
  __syncthreads();
  for (int off = 1; off < NCOARSE; off <<= 1) {
    uint32_t v = scan[tid];
    uint32_t a = (tid + off < NCOARSE) ? scan[tid + off] : 0u;
    __syncthreads();
    scan[tid] = v + a;
    __syncthreads();
  }
  {
    uint32_t sj  = scan[tid];
    uint32_t sj1 = (tid + 1 < NCOARSE) ? scan[tid + 1] : 0u;
    if (sj >= (uint32_t)kk && sj1 < (uint32_t)kk) { sC = tid; sAbove = sj1; }
  }
  __syncthreads();
  if (tid == 0) {
    const int c = sC;
    uint32_t above = sAbove;
    int t = c * 8;
    for (int b = c * 8 + 7; b >= c * 8; --b) {
      uint32_t h = hist[b];
      if (above + h >= (uint32_t)kk) { t = b; break; }
      above += h;
    }
    sTbin = t;
  }
  __syncthreads();
  const uint32_t tbin = (uint32_t)sTbin;

  // ---- Pass 2: collect candidates (row is still L2-resident: 65MB<192MB) --
  const float4* __restrict__ rp4 = (const float4*)rp;
  const int V4 = V >> 2;
  for (int i = tid; i < V4; i += BDIM) {
    float4 v = rp4[i];
    float vv[4] = {v.x, v.y, v.z, v.w};
#pragma unroll
    for (int e = 0; e < 4; ++e) {
      uint32_t mk = mono_key(vv[e]);
      if ((mk >> 20) >= tbin) {
        uint32_t pos = atomicAdd(&ncand, 1u);
        if (pos < CANDMAX) {
          uint32_t idx = (uint32_t)(i * 4 + e);
          // sort key: value desc, index asc on ties (matches top_k tie-break)
          cand[pos] = ((unsigned long long)mk << 32) |
                      (unsigned long long)(0xFFFFFFFFu - idx);
        }
      }
    }
  }
  __syncthreads();
  uint32_t n = ncand; if (n > CANDMAX) n = CANDMAX;
  for (int i = tid; i < CANDMAX; i += BDIM)
    if ((uint32_t)i >= n) cand[i] = 0ull;            // pad sinks to the bottom
  __syncthreads();

  // ---- Bitonic sort (descending) over CANDMAX u64 keys --------------------
  for (unsigned k2 = 2; k2 <= CANDMAX; k2 <<= 1) {
    for (unsigned j = k2 >> 1; j > 0; j >>= 1) {
      for (unsigned t = tid; t < CANDMAX; t += BDIM) {
        unsigned l = t ^ j;
        if (l > t) {
          unsigned long long a = cand[t], b = cand[l];
          bool up = ((t & k2) == 0);                 // "up" block => descending
          bool sw = up ? (a < b) : (a > b);
          if (sw) { cand[t] = b; cand[l] = a; }
        }
      }
      __syncthreads();
    }
  }

  // ---- Softmax / top-p / sample (tiny; thread 0) ---------------------------
  if (tid == 0) {
    int neff = (int)n; if (neff > kk) neff = kk; if (neff < 1) neff = 1;
    float scaled[KMAX]; uint32_t idxs[KMAX]; float probs[KMAX]; bool inc[KMAX];
    for (int i = 0; i < neff; ++i) {
      unsigned long long c = cand[i];
      scaled[i] = mono_inv((uint32_t)(c >> 32)) * 1.25f;   // logits / 0.8
      idxs[i]   = 0xFFFFFFFFu - (uint32_t)c;
    }
    float m = scaled[0], sum = 0.f;
    for (int i = 0; i < neff; ++i) { probs[i] = __expf(scaled[i] - m); sum += probs[i]; }
    float inv = 1.f / sum, cdf = 0.f, fsum = 0.f;
    for (int i = 0; i < neff; ++i) {
      float p = probs[i] * inv;
      inc[i] = (i == 0) || (cdf < 0.9f);   // shifted mask, first always kept
      cdf += p;
      probs[i] = p;
      if (inc[i]) fsum += p;
    }
    float u = u01((uint32_t)row) * fsum;
    float acc = 0.f; int pick = 0;
    for (int i = 0; i < neff; ++i) {
      if (!inc[i]) continue;
      acc += probs[i]; pick = i;
      if (acc >= u) break;
    }
    out[row]     = scaled[pick];           // final_logits
    out[B + row] = (float)idxs[pick];      // next_token (as float)
  }
}

extern "C" void kernel_launch(void* const* d_in, const int* in_sizes, int n_in,
                              void* d_out, int out_size, void* d_ws, size_t ws_size,
                              hipStream_t stream) {
  (void)n_in; (void)d_ws; (void)ws_size; (void)out_size;
  const float* logits = (const float*)d_in[0];
  const int*   kptr   = (const int*)d_in[1];
  float*       out    = (float*)d_out;
  const int V = 128000;
  const int B = in_sizes[0] / V;
  Sampler_40836549050652_kernel<<<dim3(B), dim3(BDIM), 0, stream>>>(
      logits, kptr, out, B, V);
}